// GCN_11965778886828
// MI455X (gfx1250) — compile-verified
//
#include <hip/hip_runtime.h>
#include <math.h>

typedef __attribute__((ext_vector_type(2))) float v2f;
typedef __attribute__((ext_vector_type(8))) float v8f;

#define NEG_SLOPE 0.2f

// Problem constants (match reference)
#define BATCH 32
#define FBAND 128
#define TLEN  2048
#define HID   256
#define NCLS  64
#define AWIN  15
#define NODE  128
#define BA    (BATCH * AWIN)   // 480

static __device__ __forceinline__ v8f wmma4(v2f a, v2f b, v8f c) {
  // D = A(16x4,f32) * B(4x16,f32) + C(16x16,f32)
  return __builtin_amdgcn_wmma_f32_16x16x4_f32(false, a, false, b, (short)0, c,
                                               false, false);
}

// ---------------------------------------------------------------------------
// Generic batched WMMA-f32 GEMM: C = op(A @ B + bias), tile = 128(M) x 64(N)
// 256 threads = 8 waves; wave w owns a 16-row strip, 4 col-tiles of 16.
// A row-major [M,K] (lda), B row-major [K,N] (ldb), C row-major (ldc).
// Batch offsets: A += z*sA ; B += (z/bDiv)*sB ; C += z*sC.
// Staging: register-buffered (many loads in flight), B packed as K-pairs
// so each WMMA B operand is a single ds_load_b64.
// ---------------------------------------------------------------------------
#define TK 32
__global__ __launch_bounds__(256) void wmma_gemm_f32(
    const float* __restrict__ A, const float* __restrict__ B,
    const float* __restrict__ bias, float* __restrict__ C,
    int K,
    long long sA, long long sB, long long sC, int bDiv,
    int lda, int ldb, int ldc, int leaky)
{
  __shared__ float As[TK][129];        // [k][m], padded (odd -> conflict-free)
  __shared__ float Bs[TK / 2][64][2];  // [k/2][n][k&1]: v2f operands contiguous

  const int z  = blockIdx.z;
  const int m0 = blockIdx.y * 128;
  const int n0 = blockIdx.x * 64;
  const float* Ab = A + (long long)z * sA;
  const float* Bb = B + (long long)(z / bDiv) * sB;
  float*       Cb = C + (long long)z * sC;

  const int t    = threadIdx.x;
  const int lane = t & 31;
  const int wave = t >> 5;
  const int r0   = wave * 16;
  const int lm   = lane & 15;
  const int lh   = lane >> 4;

  // staging coordinates
  const int am = t >> 5;               // A: starting row (step 8), 0..7
  const int ak = t & 31;               // A: k within tile
  const int bk = t >> 6;               // B: starting k (step 4), 0..3
  const int bn = t & 63;               // B: column within tile
  const long long ldaStep = (long long)8 * lda;
  const long long ldbStep = (long long)4 * ldb;
  const float* aPtr0 = Ab + (long long)(m0 + am) * lda + ak;
  const float* bPtr0 = Bb + (long long)bk * ldb + (n0 + bn);

  v8f acc[4];
  #pragma unroll
  for (int c = 0; c < 4; ++c)
    #pragma unroll
    for (int r = 0; r < 8; ++r) acc[c][r] = 0.0f;

  for (int k0 = 0; k0 < K; k0 += TK) {
    // register-buffered staging: all loads issued before any store/wait
    float ra[16];
    {
      const float* p = aPtr0 + k0;
      #pragma unroll
      for (int i = 0; i < 16; ++i) { ra[i] = *p; p += ldaStep; }
    }
    float rb[8];
    {
      const float* p = bPtr0 + (long long)k0 * ldb;
      #pragma unroll
      for (int i = 0; i < 8; ++i) { rb[i] = *p; p += ldbStep; }
    }
    __syncthreads();
    #pragma unroll
    for (int i = 0; i < 16; ++i) As[ak][am + 8 * i] = ra[i];
    #pragma unroll
    for (int i = 0; i < 8; ++i) Bs[(bk >> 1) + 2 * i][bn][bk & 1] = rb[i];
    __syncthreads();

    #pragma unroll
    for (int kk = 0; kk < TK; kk += 4) {
      v2f a;
      a[0] = As[kk + 2 * lh][r0 + lm];
      a[1] = As[kk + 2 * lh + 1][r0 + lm];
      const int q = (kk + 2 * lh) >> 1;     // even K-pair index
      #pragma unroll
      for (int c = 0; c < 4; ++c) {
        v2f b = *(const v2f*)&Bs[q][c * 16 + lm][0];  // single b64 load
        acc[c] = wmma4(a, b, acc[c]);
      }
    }
  }

  #pragma unroll
  for (int c = 0; c < 4; ++c) {
    int col = n0 + c * 16 + lm;
    float bv = bias ? bias[col] : 0.0f;
    #pragma unroll
    for (int r = 0; r < 8; ++r) {
      int row = m0 + r0 + r + 8 * lh;
      float v = acc[c][r] + bv;
      if (leaky) v = (v >= 0.0f) ? v : NEG_SLOPE * v;
      Cb[(long long)row * ldc + col] = v;
    }
  }
}

// ---------------------------------------------------------------------------
// adj[b,a] = nan_to_num((clip(corrcoef(window.T), -1, 1) + 1) * 0.5)
// One block per (b,a). m[n][f] = x[b, f, a*128 + n]; Gram via WMMA f32.
// Both WMMA operands read contiguous [row][f..f+1] pairs (single b64 loads).
// ---------------------------------------------------------------------------
__global__ __launch_bounds__(256) void corr_adj_kernel(
    const float* __restrict__ x, float* __restrict__ adj)
{
  __shared__ float mS[NODE][NODE + 2];  // centered m, padded (even pad keeps
                                        // 8B alignment for b64 pair loads)
  __shared__ float dS[NODE];

  const int ba = blockIdx.x;
  const int b = ba / AWIN, a = ba % AWIN;
  const float* xb = x + (long long)b * FBAND * TLEN + (long long)a * NODE;

  const int t = threadIdx.x;
  // register-buffered staging in chunks of 8 (keeps 8 loads in flight)
  const int f0 = t >> 7;                // starting f (step 2)
  const int n  = t & 127;
  const float* xp = xb + (long long)f0 * TLEN + n;
  #pragma unroll
  for (int blk = 0; blk < 64; blk += 8) {
    float r[8];
    #pragma unroll
    for (int j = 0; j < 8; ++j)
      r[j] = xp[(long long)(2 * (blk + j)) * TLEN];
    #pragma unroll
    for (int j = 0; j < 8; ++j)
      mS[n][f0 + 2 * (blk + j)] = r[j];
  }
  __syncthreads();
  if (t < NODE) {
    float s = 0.0f;
    for (int f = 0; f < NODE; ++f) s += mS[t][f];
    float mean = s * (1.0f / NODE);
    float d2 = 0.0f;
    for (int f = 0; f < NODE; ++f) {
      float v = mS[t][f] - mean;
      mS[t][f] = v;
      d2 += v * v;
    }
    dS[t] = sqrtf(fmaxf(d2, 0.0f));
  }
  __syncthreads();

  const int lane = t & 31, wave = t >> 5;
  const int r0 = wave * 16, lm = lane & 15, lh = lane >> 4;
  v8f acc[8];
  #pragma unroll
  for (int c = 0; c < 8; ++c)
    #pragma unroll
    for (int r = 0; r < 8; ++r) acc[c][r] = 0.0f;

  for (int f = 0; f < NODE; f += 4) {
    v2f av = *(const v2f*)&mS[r0 + lm][f + 2 * lh];
    #pragma unroll
    for (int c = 0; c < 8; ++c) {         // B = m^T : B[K=f][N=j] = m[j][f]
      v2f bv = *(const v2f*)&mS[c * 16 + lm][f + 2 * lh];
      acc[c] = wmma4(av, bv, acc[c]);
    }
  }

  float* adjb = adj + (long long)ba * NODE * NODE;
  #pragma unroll
  for (int c = 0; c < 8; ++c) {
    int j = c * 16 + lm;
    float dj = dS[j];
    #pragma unroll
    for (int r = 0; r < 8; ++r) {
      int i = r0 + r + 8 * lh;
      float rr = acc[c][r] / (dS[i] * dj);
      float o;
      if (rr != rr) {                     // NaN -> 0 (nan_to_num)
        o = 0.0f;
      } else {
        rr = rr > 1.0f ? 1.0f : (rr < -1.0f ? -1.0f : rr);
        o = (rr + 1.0f) * 0.5f;
      }
      adjb[(long long)i * NODE + j] = o;
    }
  }
}

// xa[ba][c] = mean over nodes of XApre[ba][n][c]
__global__ void mean_nodes_kernel(const float* __restrict__ xap,
                                  float* __restrict__ xa)
{
  int ba = blockIdx.x, c = threadIdx.x;   // 64 threads
  const float* p = xap + (long long)ba * NODE * NCLS;
  float s = 0.0f;
  for (int n = 0; n < NODE; ++n) s += p[n * NCLS + c];
  xa[ba * NCLS + c] = s * (1.0f / NODE);
}

// adj2[b] = clip(corrcoef(xa[b]), -1, 1)  (NaN preserved, as in reference)
__global__ __launch_bounds__(256) void corr2_kernel(
    const float* __restrict__ xa, float* __restrict__ adj2)
{
  __shared__ float ms[AWIN][NCLS];
  __shared__ float d[AWIN];
  int b = blockIdx.x, t = threadIdx.x;
  const float* xb = xa + (long long)b * AWIN * NCLS;
  for (int idx = t; idx < AWIN * NCLS; idx += 256)
    ms[idx / NCLS][idx % NCLS] = xb[idx];
  __syncthreads();
  if (t < AWIN) {
    float s = 0.0f;
    for (int k = 0; k < NCLS; ++k) s += ms[t][k];
    float mean = s * (1.0f / NCLS);
    float d2 = 0.0f;
    for (int k = 0; k < NCLS; ++k) {
      float v = ms[t][k] - mean; ms[t][k] = v; d2 += v * v;
    }
    d[t] = sqrtf(fmaxf(d2, 0.0f));
  }
  __syncthreads();
  for (int idx = t; idx < AWIN * AWIN; idx += 256) {
    int i = idx / AWIN, j = idx % AWIN;
    float s = 0.0f;
    for (int k = 0; k < NCLS; ++k) s += ms[i][k] * ms[j][k];
    float rr = s / (d[i] * d[j]);
    rr = rr > 1.0f ? 1.0f : (rr < -1.0f ? -1.0f : rr);  // NaN falls through
    adj2[(long long)b * AWIN * AWIN + idx] = rr;
  }
}

// out[b] = max over windows of leaky(adj2 @ xa @ W4 + b4)
__global__ __launch_bounds__(256) void head_kernel(
    const float* __restrict__ xa, const float* __restrict__ adj2,
    const float* __restrict__ W4, const float* __restrict__ b4,
    float* __restrict__ out)
{
  __shared__ float xs[AWIN][NCLS];
  __shared__ float a2[AWIN][AWIN];
  __shared__ float ys[AWIN][NCLS];
  int b = blockIdx.x, t = threadIdx.x;
  const float* xb = xa + (long long)b * AWIN * NCLS;
  for (int idx = t; idx < AWIN * NCLS; idx += 256)
    xs[idx / NCLS][idx % NCLS] = xb[idx];
  for (int idx = t; idx < AWIN * AWIN; idx += 256)
    a2[idx / AWIN][idx % AWIN] = adj2[(long long)b * AWIN * AWIN + idx];
  __syncthreads();
  for (int idx = t; idx < AWIN * NCLS; idx += 256) {
    int i = idx / NCLS, c = idx % NCLS;
    float s = 0.0f;
    for (int j = 0; j < AWIN; ++j) s += a2[i][j] * xs[j][c];
    ys[i][c] = s;
  }
  __syncthreads();
  for (int idx = t; idx < AWIN * NCLS; idx += 256) {
    int i = idx / NCLS, c = idx % NCLS;
    float s = b4[c];
    for (int h = 0; h < NCLS; ++h) s += ys[i][h] * W4[h * NCLS + c];
    xs[i][c] = (s >= 0.0f) ? s : NEG_SLOPE * s;   // reuse xs for z
  }
  __syncthreads();
  if (t < NCLS) {
    float best = xs[0][t];
    for (int i = 1; i < AWIN; ++i) {
      float v = xs[i][t];
      best = (v > best) ? v : best;
    }
    out[b * NCLS + t] = best;
  }
}

// ---------------------------------------------------------------------------
extern "C" void kernel_launch(void* const* d_in, const int* in_sizes, int n_in,
                              void* d_out, int out_size, void* d_ws, size_t ws_size,
                              hipStream_t stream) {
  (void)in_sizes; (void)n_in; (void)out_size; (void)ws_size;
  const float* x  = (const float*)d_in[0];   // [32,128,2048]
  const float* W1 = (const float*)d_in[1];   // [2048,256]
  const float* b1 = (const float*)d_in[2];
  const float* W2 = (const float*)d_in[3];   // [256,256]
  const float* b2 = (const float*)d_in[4];
  const float* W3 = (const float*)d_in[5];   // [256,64]
  const float* b3 = (const float*)d_in[6];
  const float* W4 = (const float*)d_in[7];   // [64,64]
  const float* b4 = (const float*)d_in[8];
  float* out = (float*)d_out;                // [32,64]

  float* ws = (float*)d_ws;
  // workspace layout (floats)
  float* adj   = ws;                                   // 480*128*128 = 7,864,320
  float* P1    = adj  + (long long)BA * NODE * NODE;   // 32*128*256  = 1,048,576
  float* buf1  = P1   + (long long)BATCH * NODE * HID; // H1 then H2  = 15,728,640
  float* buf2  = buf1 + (long long)BA * NODE * HID;    // P2, then P3+XApre
  float* xa    = buf2 + (long long)BA * NODE * HID;    // 480*64
  float* adj2  = xa   + (long long)BA * NCLS;          // 32*225
  float* P3    = buf2;                                 // 480*128*64
  float* XApre = buf2 + (long long)BA * NODE * NCLS;   // 480*128*64

  // 1) P1 = x @ W1                       [B,128,2048] @ [2048,256]
  wmma_gemm_f32<<<dim3(HID / 64, 1, BATCH), 256, 0, stream>>>(
      x, W1, nullptr, P1, TLEN,
      (long long)NODE * TLEN, 0LL, (long long)NODE * HID, 1,
      TLEN, HID, HID, 0);

  // 2) adj = nan_to_num((corrcoef + 1) / 2) per (b,a)
  corr_adj_kernel<<<BA, 256, 0, stream>>>(x, adj);

  // 3) H1 = leaky(adj @ P1[b] + b1)      buf1
  wmma_gemm_f32<<<dim3(HID / 64, 1, BA), 256, 0, stream>>>(
      adj, P1, b1, buf1, NODE,
      (long long)NODE * NODE, (long long)NODE * HID, (long long)NODE * HID, AWIN,
      NODE, HID, HID, 1);

  // 4) P2 = H1 @ W2                      [61440,256] @ [256,256] -> buf2
  wmma_gemm_f32<<<dim3(HID / 64, BA, 1), 256, 0, stream>>>(
      buf1, W2, nullptr, buf2, HID,
      0LL, 0LL, 0LL, 1,
      HID, HID, HID, 0);

  // 5) H2 = leaky(adj @ P2 + b2)         -> buf1 (overwrites H1)
  wmma_gemm_f32<<<dim3(HID / 64, 1, BA), 256, 0, stream>>>(
      adj, buf2, b2, buf1, NODE,
      (long long)NODE * NODE, (long long)NODE * HID, (long long)NODE * HID, 1,
      NODE, HID, HID, 1);

  // 6) P3 = H2 @ W3                      [61440,256] @ [256,64] -> buf2 head
  wmma_gemm_f32<<<dim3(NCLS / 64, BA, 1), 256, 0, stream>>>(
      buf1, W3, nullptr, P3, HID,
      0LL, 0LL, 0LL, 1,
      HID, NCLS, NCLS, 0);

  // 7) XApre = leaky(adj @ P3 + b3)      -> buf2 tail
  wmma_gemm_f32<<<dim3(NCLS / 64, 1, BA), 256, 0, stream>>>(
      adj, P3, b3, XApre, NODE,
      (long long)NODE * NODE, (long long)NODE * NCLS, (long long)NODE * NCLS, 1,
      NODE, NCLS, NCLS, 1);

  // 8) xa = mean over nodes
  mean_nodes_kernel<<<BA, NCLS, 0, stream>>>(XApre, xa);

  // 9) adj2 = clip(corrcoef(xa), -1, 1)
  corr2_kernel<<<BATCH, 256, 0, stream>>>(xa, adj2);

  // 10) out = max_A leaky(adj2 @ xa @ W4 + b4)
  head_kernel<<<BATCH, 256, 0, stream>>>(xa, adj2, W4, b4, out);
}